// QsrDecoder_40501541601486
// MI455X (gfx1250) — compile-verified
//
#include <hip/hip_runtime.h>
#include <math.h>

#define BB 4
#define TT 4
#define NN 32
#define CC 8
#define HH 128
#define WW 128
#define HWPIX (HH*WW)
#define NEG_INF_F (-1e9f)
#define EPS_F (1e-8f)

typedef __attribute__((ext_vector_type(2))) float v2f;
typedef __attribute__((ext_vector_type(8))) float v8f;

__device__ __forceinline__ float fast_rcp(float x) {
    return __builtin_amdgcn_rcpf(x);          // v_rcp_f32, ~1 ulp
}
__device__ __forceinline__ float sigmoidf_fast(float x) {
    return fast_rcp(1.0f + __expf(-x));       // v_exp_f32 + v_rcp_f32
}

// ---------------------------------------------------------------------------
// Kernel 1: shapes[bt,n,h,w] = sigmoid(min_c cfield) * valid, cfield via WMMA.
// cfield[(n,c), p] = [s*nx, s*ny, s*bias', 0] . [gx, gy, 1, 0]
// One v_wmma_f32_16x16x4_f32 = 16 (n,c) rows (2 nodes x 8 constraints)
// x 16 pixels. D layout: lanes 0-15 -> node0 (c in v0..v7), lanes 16-31 ->
// node1; so min over C is a per-lane min over the 8 accumulator VGPRs.
// grid.x = bts*16 (pixel chunks of 1024), block = 256 (8 waves).
// ---------------------------------------------------------------------------
__global__ __launch_bounds__(256)
void qsr_shapes_wmma(const float* __restrict__ shape_inputs,
                     const float* __restrict__ hw,
                     const float* __restrict__ valid,
                     float* __restrict__ shapes_ws,
                     int bt_start)
{
    const int local_bt = blockIdx.x >> 4;
    const int chunk    = blockIdx.x & 15;       // 1024-pixel chunk
    const int bt       = bt_start + local_bt;
    const int lane     = threadIdx.x & 31;
    const int wid      = threadIdx.x >> 5;      // 0..7
    const int half     = lane >> 4;             // K-half selector
    const int r        = lane & 15;             // A row / B col

    #pragma unroll
    for (int pp = 0; pp < 2; ++pp) {
        const int pair = wid + pp * 8;          // 0..15 (node pair)

        // ---- A matrix: row r = (node = pair*2 + r/8, constraint c = r%8)
        const int nA = pair * 2 + (r >> 3);
        const int c  = r & 7;
        const float4 code = *reinterpret_cast<const float4*>(
            shape_inputs + ((size_t)(bt * NN + nA)) * (4 * CC) + c * 4);
        // xy_scales = [63.5, 63.5, 1.0, 63.5]
        const float nx   = code.x * 63.5f;
        const float ny   = code.y * 63.5f;
        const float bias = code.z;
        const float sc   = code.w * 63.5f;
        const float chn  = hw[(bt * NN + nA) * 2 + 0];
        const float cwn  = hw[(bt * NN + nA) * 2 + 1];
        const float txp  =  cwn * 63.5f;
        const float typ  = -chn * 63.5f;
        const float k0 = sc * nx;
        const float k1 = sc * ny;
        const float k2 = sc * (bias - nx * txp - ny * typ);
        v2f a;
        a.x = half ? k2 : k0;   // VGPR0: K0 (lanes 0-15) / K2 (lanes 16-31)
        a.y = half ? 0.0f : k1; // VGPR1: K1 / K3

        // ---- output-side mapping: lane half selects node within the pair
        const int   n_out = pair * 2 + half;
        const float vld   = valid[bt * NN + n_out];
        float* outp = shapes_ws + ((size_t)(local_bt * NN + n_out)) * HWPIX;

        for (int t = 0; t < 64; ++t) {
            const int p0 = chunk * 1024 + t * 16;
            const int p  = p0 + r;
            const int ph = p >> 7;
            const int pw = p & 127;
            const float gx = (float)pw - 63.5f;   // xs[w] = w - (W-1)/2
            const float gy = 63.5f - (float)ph;   // ys[h] = -(h - (H-1)/2)
            v2f b;
            b.x = half ? 1.0f : gx;
            b.y = half ? 0.0f : gy;
            v8f d = {};
            d = __builtin_amdgcn_wmma_f32_16x16x4_f32(
                    false, a, false, b, (short)0, d, false, false);
            float mn = fminf(fminf(fminf(d[0], d[1]), fminf(d[2], d[3])),
                             fminf(fminf(d[4], d[5]), fminf(d[6], d[7])));
            outp[p0 + r] = sigmoidf_fast(mn) * vld;
        }
    }
}

// ---------------------------------------------------------------------------
// Kernel 2: fused 5x5 conv (depth/delta_h/delta_w analytic, shape from LDS
// halo tile) + log-mask logits + online softmax over N + attribute blend.
// grid.x = bts*64 (8x8 tiles of 16x16 pixels), block = 256.
// ---------------------------------------------------------------------------
__global__ __launch_bounds__(256)
void qsr_render(const float* __restrict__ shapes_ws,
                const float* __restrict__ depths,
                const float* __restrict__ hw,
                const float* __restrict__ valid,
                const float* __restrict__ attrs,
                const float* __restrict__ conv_w,
                const float* __restrict__ conv_b,
                float* __restrict__ out,
                int bt_start)
{
    __shared__ float s_tile[20 * 20];   // shape halo (zero = SAME padding)
    __shared__ float s_w[25][4];        // conv weights [tap][ci]
    __shared__ float s_par[NN][8];      // depth, ch, cw, valid, a0, a1, a2

    const int local_bt = blockIdx.x >> 6;
    const int tile     = blockIdx.x & 63;
    const int bt       = bt_start + local_bt;
    const int h0 = (tile >> 3) * 16;
    const int w0 = (tile & 7) * 16;
    const int tid = threadIdx.x;
    const int ty = tid >> 4, tx = tid & 15;
    const int h = h0 + ty, w = w0 + tx;

    if (tid < NN) {
        const int gi = bt * NN + tid;
        s_par[tid][0] = depths[gi];
        s_par[tid][1] = hw[gi * 2 + 0];
        s_par[tid][2] = hw[gi * 2 + 1];
        s_par[tid][3] = valid[gi];
        s_par[tid][4] = attrs[gi * 3 + 0];
        s_par[tid][5] = attrs[gi * 3 + 1];
        s_par[tid][6] = attrs[gi * 3 + 2];
    } else if (tid < NN + 100) {
        const int i = tid - NN;         // ((kh*5+kw)*4 + ci)
        s_w[i >> 2][i & 3] = conv_w[i];
    }
    const float cb = conv_b[0];

    // online softmax state
    float m = -1e30f, ssum = 0.0f, ac0 = 0.0f, ac1 = 0.0f, ac2 = 0.0f;

    for (int n = 0; n < NN; ++n) {
        __syncthreads();   // previous tile fully consumed
        const float* sp = shapes_ws + ((size_t)(local_bt * NN + n)) * HWPIX;
        for (int i = tid; i < 400; i += 256) {
            const int rr = i / 20, cc2 = i % 20;
            const int hh = h0 + rr - 2, wb = w0 + cc2 - 2;
            float v = 0.0f;
            if (hh >= 0 && hh < HH && wb >= 0 && wb < WW)
                v = sp[hh * WW + wb];
            s_tile[i] = v;
        }
        __syncthreads();

        const float depth = s_par[n][0], chn = s_par[n][1];
        const float cwn   = s_par[n][2], vld = s_par[n][3];

        float cacc = cb;
        #pragma unroll
        for (int kh = 0; kh < 5; ++kh) {
            const int   hh  = h + kh - 2;
            const float hgv = -1.0f + (float)hh * (2.0f / 127.0f);
            const bool  hin = (hh >= 0) && (hh < HH);
            #pragma unroll
            for (int kw = 0; kw < 5; ++kw) {
                const int   wb  = w + kw - 2;
                const float wgv = -1.0f + (float)wb * (2.0f / 127.0f);
                const float ib  = (hin && wb >= 0 && wb < WW) ? 1.0f : 0.0f;
                const float sv  = s_tile[(ty + kh) * 20 + (tx + kw)];
                const float* wv = s_w[kh * 5 + kw];
                cacc += ib * (wv[0] * depth + wv[1] * (hgv - chn)
                                            + wv[2] * (wgv - cwn))
                        + wv[3] * sv;
            }
        }

        const float scen  = s_tile[(ty + 2) * 20 + (tx + 2)];
        float logit = __logf(scen + EPS_F) - cacc;
        logit = (vld > 0.0f) ? logit : NEG_INF_F;

        const float a0 = s_par[n][4], a1 = s_par[n][5], a2 = s_par[n][6];
        if (logit > m) {
            const float rsc = __expf(m - logit);
            ssum = ssum * rsc + 1.0f;
            ac0 = ac0 * rsc + a0;
            ac1 = ac1 * rsc + a1;
            ac2 = ac2 * rsc + a2;
            m = logit;
        } else {
            const float e = __expf(logit - m);
            ssum += e;
            ac0 += e * a0;
            ac1 += e * a1;
            ac2 += e * a2;
        }
    }

    const float inv = fast_rcp(ssum);
    float* op = out + ((size_t)(bt * HH + h) * WW + w) * 3;
    op[0] = ac0 * inv;
    op[1] = ac1 * inv;
    op[2] = ac2 * inv;
}

// ---------------------------------------------------------------------------
extern "C" void kernel_launch(void* const* d_in, const int* in_sizes, int n_in,
                              void* d_out, int out_size, void* d_ws, size_t ws_size,
                              hipStream_t stream) {
    const float* shape_inputs = (const float*)d_in[0];
    const float* depthsp      = (const float*)d_in[1];
    const float* hwp          = (const float*)d_in[2];
    const float* validp       = (const float*)d_in[3];
    const float* attrsp       = (const float*)d_in[4];
    const float* convw        = (const float*)d_in[5];
    const float* convb        = (const float*)d_in[6];
    float*       outp         = (float*)d_out;
    float*       ws           = (float*)d_ws;

    const int    BT     = BB * TT;                            // 16
    const size_t per_bt = (size_t)NN * HWPIX * sizeof(float); // 2 MiB
    int maxbts = (int)(ws_size / per_bt);
    if (maxbts < 1)  maxbts = 1;   // best effort
    if (maxbts > BT) maxbts = BT;

    for (int bt0 = 0; bt0 < BT; bt0 += maxbts) {
        const int bts = (BT - bt0 < maxbts) ? (BT - bt0) : maxbts;
        qsr_shapes_wmma<<<dim3(bts * 16), dim3(256), 0, stream>>>(
            shape_inputs, hwp, validp, ws, bt0);
        qsr_render<<<dim3(bts * 64), dim3(256), 0, stream>>>(
            ws, depthsp, hwp, validp, attrsp, convw, convb, outp, bt0);
    }
}